// QuanvolutionFilter_65481071408081
// MI455X (gfx1250) — compile-verified
//
#include <hip/hip_runtime.h>

typedef float v2f __attribute__((ext_vector_type(2)));
typedef float v8f __attribute__((ext_vector_type(8)));

#define N_STATES (4096 * 196)           // 802816 product states
static constexpr float HALF_PI = 1.57079632679489662f;

// ---------------------------------------------------------------------------
// Build the fixed 16x16 complex unitary U from the 36 variational params.
// One block of 16 threads; thread t owns column t of U in registers.
// Layout in ws: uc[0..255] = Re(U) row-major, uc[256..511] = Im(U) row-major.
// ---------------------------------------------------------------------------
__global__ __launch_bounds__(16) void build_unitary_kernel(
    const float* __restrict__ params, float* __restrict__ uc) {
  const int t = threadIdx.x;
  if (t >= 16) return;

  float re[16], im[16];
#pragma unroll
  for (int r = 0; r < 16; ++r) { re[r] = (r == t) ? 1.0f : 0.0f; im[r] = 0.0f; }

  int off = 0;
  for (int layer = 0; layer < 3; ++layer) {
    for (int w = 0; w < 4; ++w) {
      const int p = 3 - w;               // big-endian: qubit w is bit (3-w)
      for (int g = 0; g < 3; ++g) {      // RY, RZ, RY
        float ch, sh;
        __sincosf(0.5f * params[off++], &sh, &ch);
        if (g == 1) {
          // RZ: rows with bit clear *= (ch - i*sh); bit set *= (ch + i*sh)
#pragma unroll
          for (int r0 = 0; r0 < 16; ++r0) {
            const float a = re[r0], b = im[r0];
            if (r0 & (1 << p)) { re[r0] = ch * a - sh * b; im[r0] = ch * b + sh * a; }
            else               { re[r0] = ch * a + sh * b; im[r0] = ch * b - sh * a; }
          }
        } else {
          // RY: row0' = c*row0 - s*row1 ; row1' = s*row0 + c*row1
#pragma unroll
          for (int r0 = 0; r0 < 16; ++r0) {
            if (r0 & (1 << p)) continue;
            const int r1 = r0 | (1 << p);
            const float a0 = re[r0], b0 = im[r0], a1 = re[r1], b1 = im[r1];
            re[r0] = ch * a0 - sh * a1;  im[r0] = ch * b0 - sh * b1;
            re[r1] = sh * a0 + ch * a1;  im[r1] = sh * b0 + ch * b1;
          }
        }
      }
    }
    // Entangler: CNOT(0,1), CNOT(1,2), CNOT(2,3), CNOT(3,0) — row swaps
#pragma unroll
    for (int e = 0; e < 4; ++e) {
      const int cw = e, tw = (e + 1) & 3;
      const int pc = 3 - cw, pt = 3 - tw;
#pragma unroll
      for (int r = 0; r < 16; ++r) {
        if ((r & (1 << pc)) && !(r & (1 << pt))) {
          const int r2 = r | (1 << pt);
          float a = re[r]; re[r] = re[r2]; re[r2] = a;
          float b = im[r]; im[r] = im[r2]; im[r2] = b;
        }
      }
    }
  }
#pragma unroll
  for (int r = 0; r < 16; ++r) {
    uc[r * 16 + t]       = re[r];
    uc[256 + r * 16 + t] = im[r];
  }
}

// ---------------------------------------------------------------------------
// Main kernel: one wave handles a tile of 16 states.
//   psi^T[16 basis x 16 states] = U[16x16] @ psi0^T[16 x 16 states]
// done as 4 chained V_WMMA_F32_16X16X4_F32 per Re/Im (A = U slice, constant;
// B = psi0 slice built on the fly from cos/sin products).
// D layout: VGPR r, lanes 0-15 -> (j=r, n=lane); lanes 16-31 -> (j=r+8).
// ZS signs are bits of j -> Walsh combine over r + one shfl_xor(16).
// ---------------------------------------------------------------------------
__global__ __launch_bounds__(256) void quanv_kernel(
    const float* __restrict__ x, const float* __restrict__ uc,
    float* __restrict__ out) {
  const int lane = threadIdx.x & 31;
  const int h = lane >> 4;    // lane half: selects K pair (A) / qubit-2 bit (B)
  const int m = lane & 15;    // A row index == state index within tile

  // A operands: Re/Im of U, 16x4 f32 slices. Lane L<16: A[m, 4s+{0,1}],
  // lanes >=16: A[m, 4s+{2,3}]  (ISA 16x4 f32 A layout).
  v2f Are[4], Aim[4];
#pragma unroll
  for (int s = 0; s < 4; ++s) {
    const int idx = m * 16 + 4 * s + 2 * h;
    Are[s] = *(const v2f*)(uc + idx);
    Aim[s] = *(const v2f*)(uc + 256 + idx);
  }

  const int nTiles = N_STATES / 16;
  const int wave   = (int)((blockIdx.x * blockDim.x + threadIdx.x) >> 5);
  const int nWaves = (int)((gridDim.x * blockDim.x) >> 5);

  for (int tile = wave; tile < nTiles; tile += nWaves) {
    const int n = tile * 16 + m;          // global state id for this lane
    // decode patch: n = b*196 + r*14 + c ; pixels at b*784 + 56r + 2c + {0,1,28,29}
    const int b = n / 196;
    const int p = n - b * 196;
    const int r = p / 14;
    const int c = p - r * 14;
    const float* px = x + b * 784 + r * 56 + c * 2;
    const float t0 = px[0], t1 = px[1], t2 = px[28], t3 = px[29];

    float c0, s0, c1, s1, c2, s2, c3, s3;
    __sincosf(t0 * HALF_PI, &s0, &c0);    // half-angle: patch * pi / 2
    __sincosf(t1 * HALF_PI, &s1, &c1);
    __sincosf(t2 * HALF_PI, &s2, &c2);
    __sincosf(t3 * HALF_PI, &s3, &c3);

    // B slice s supplies psi0[k], k = 4s + v + 2h: bit1(k)=qubit2=h, bit0=qubit3=v
    const float q2 = h ? s2 : c2;
    float ms[4];
    ms[0] = c0 * c1 * q2;                 // (a,b) = bits of s -> qubit0/qubit1
    ms[1] = c0 * s1 * q2;
    ms[2] = s0 * c1 * q2;
    ms[3] = s0 * s1 * q2;

    v8f Dre = {0, 0, 0, 0, 0, 0, 0, 0};
    v8f Dim = {0, 0, 0, 0, 0, 0, 0, 0};
#pragma unroll
    for (int s = 0; s < 4; ++s) {
      v2f B;
      B.x = ms[s] * c3;                   // v=0 -> qubit3 = |0>
      B.y = ms[s] * s3;                   // v=1 -> qubit3 = |1>
      Dre = __builtin_amdgcn_wmma_f32_16x16x4_f32(false, Are[s], false, B,
                                                  (short)0, Dre, false, false);
      Dim = __builtin_amdgcn_wmma_f32_16x16x4_f32(false, Aim[s], false, B,
                                                  (short)0, Dim, false, false);
    }

    // probs[j=r+8h, n] and Walsh combine with ZS signs (bit(3-i) of j)
    float pr[8];
#pragma unroll
    for (int rr = 0; rr < 8; ++rr)
      pr[rr] = Dre[rr] * Dre[rr] + Dim[rr] * Dim[rr];

    const float a03 = pr[0] + pr[1] + pr[2] + pr[3];
    const float a47 = pr[4] + pr[5] + pr[6] + pr[7];
    const float e0 = a03 + a47;                                     // i=0: sign = lane half
    const float e1 = a03 - a47;                                     // i=1: bit2 of r
    const float e2 = (pr[0] + pr[1] + pr[4] + pr[5]) -
                     (pr[2] + pr[3] + pr[6] + pr[7]);               // i=2: bit1 of r
    const float e3 = (pr[0] + pr[2] + pr[4] + pr[6]) -
                     (pr[1] + pr[3] + pr[5] + pr[7]);               // i=3: bit0 of r

    const float o0 = __shfl_xor(e0, 16, 32);
    const float o1 = __shfl_xor(e1, 16, 32);
    const float o2 = __shfl_xor(e2, 16, 32);
    const float o3 = __shfl_xor(e3, 16, 32);

    if (h == 0) {                          // lanes 0-15 own the final 4 outputs
      float4 v;
      v.x = e0 - o0;                       // upper half carries j>=8 -> minus for i=0
      v.y = e1 + o1;
      v.z = e2 + o2;
      v.w = e3 + o3;
      *(float4*)(out + (size_t)n * 4) = v; // out is exactly [N,4] contiguous
    }
  }
}

extern "C" void kernel_launch(void* const* d_in, const int* in_sizes, int n_in,
                              void* d_out, int out_size, void* d_ws, size_t ws_size,
                              hipStream_t stream) {
  const float* x      = (const float*)d_in[0];   // [4096, 784]
  const float* params = (const float*)d_in[1];   // [36]
  float* out = (float*)d_out;                    // [4096, 784]
  float* uc  = (float*)d_ws;                     // 512 floats: Re(U), Im(U)

  build_unitary_kernel<<<1, 16, 0, stream>>>(params, uc);

  const int nTiles = N_STATES / 16;              // 50176 tiles, 1 wave each
  const int threads = 256;                       // 8 waves per block
  const int blocks = (nTiles + (threads / 32) - 1) / (threads / 32);  // 6272
  quanv_kernel<<<blocks, threads, 0, stream>>>(x, uc, out);
}